// SpikingConv3DLayer_11381663334466
// MI455X (gfx1250) — compile-verified
//
#include <hip/hip_runtime.h>

typedef __attribute__((ext_vector_type(2))) float v2f;
typedef __attribute__((ext_vector_type(8))) float v8f;

#define TT   50
#define NCIN 16
#define NCOUT 64
#define NH   32
#define NW   32
#define HWPOS 1024
#define KTOT 432      // NCIN*27
#define KCH  108      // K per chunk (4 cin)
#define FWSTR 436     // full-w LDS row stride (436%64=52 -> 16 rows hit 16 distinct bank-quads)
#define XSLAB 2448    // 3*6*34*4 floats per chunk slab
#define XPLANE (6*34*4) // floats per kd plane
#define SSTR 68       // padded LDS row stride, step kernel (even for b64, 68%64=4 -> conflict-free)
#define FEPS 1e-8f

// workspace layout (float offsets)
#define WS_CONV 0ull
#define WS_MEM  52428800ull                 // 16*50*64*1024 conv floats before this
#define WS_MMT  (WS_MEM + 1048576ull)       // 64*64
#define WS_INV  (WS_MMT + 4096ull)          // 64
#define WS_TSUM (WS_INV + 64ull)            // 64 (50 used)

#define NOUT_SPK 52428800ull                // B*COUT*T*H*W

// ---------------- prep: Gram, inv_norm, M^T = (1-b)v^T - b*d, zero t-sums ----
__global__ __launch_bounds__(256) void prep_kernel(const float* __restrict__ w,
                                                   const float* __restrict__ v,
                                                   const float* __restrict__ beta,
                                                   float* __restrict__ ws) {
  int tid = threadIdx.x;
  float bet = beta[0];
  float* invn = ws + WS_INV;
  float* mmt  = ws + WS_MMT;
  float* tsum = ws + WS_TSUM;
  if (tid < NCOUT) {
    const float* wr = w + (size_t)tid * KTOT;
    float s = 0.f;
    #pragma unroll 4
    for (int k = 0; k < KTOT; ++k) s += wr[k] * wr[k];
    invn[tid] = 1.0f / (s + FEPS);
    tsum[tid] = 0.f;
  }
  #pragma unroll 1
  for (int e = tid; e < NCOUT * NCOUT; e += 256) {
    int c  = e >> 6;   // output channel (row of M^T)
    int kk = e & 63;   // spike channel
    const float* wa = w + (size_t)kk * KTOT;
    const float* wb = w + (size_t)c  * KTOT;
    float dot = 0.f;
    #pragma unroll 4
    for (int k = 0; k < KTOT; ++k) dot += wa[k] * wb[k];
    mmt[c * NCOUT + kk] = (1.0f - bet) * v[kk * NCOUT + c] - bet * dot;
  }
}

// Issue async (memory -> LDS) loads for one x-slab chunk; padding lanes store 0
// via a regular DS write. Per-lane EXEC masking makes the divergence legal.
__device__ __forceinline__ void stage_x_async(const float* __restrict__ x,
                                              float* buf, int b, int t, int h0,
                                              int kc, int tid) {
  uint64_t gbase = (uint64_t)(const void*)x;
  #pragma unroll 1
  for (int li = tid; li < XSLAB; li += 256) {
    int wwi = li % 34;
    int r   = li / 34;
    int hh  = r % 6;
    int r2  = r / 6;
    int kd  = r2 % 3;
    int cl  = r2 / 3;
    int cin = kc * 4 + cl;
    int tg  = t - 1 + kd;
    int hg  = h0 - 1 + hh;
    int wg  = wwi - 1;
    float* dst = &buf[((kd * 6 + hh) * 34 + wwi) * 4 + cl];
    if (tg >= 0 && tg < TT && hg >= 0 && hg < NH && wg >= 0 && wg < NW) {
      unsigned lds_off = (unsigned)(size_t)dst;
      int goff = (((b * NCIN + cin) * TT + tg) * HWPOS + hg * NW + wg) * 4;
      asm volatile("global_load_async_to_lds_b32 %0, %1, %2"
                   :: "v"(lds_off), "v"(goff), "s"(gbase) : "memory");
    } else {
      *dst = 0.f;
    }
  }
}

// ---------------- conv: implicit GEMM, M=cout(64), N=pos, K=432 (4 chunks) ---
// Full permuted weight matrix resident in LDS (staged once); x slab double-
// buffered via async LDS loads overlapped with the WMMA work of the previous
// chunk. K-order: k' = kc*108 + q*4 + cin_local, q=(kd*3+kh)*3+kw; all tap
// loads are small immediate offsets from 7 hoisted base pointers.
__global__ __launch_bounds__(256) void conv_kernel(const float* __restrict__ x,
                                                   const float* __restrict__ w,
                                                   const float* __restrict__ beta,
                                                   float* __restrict__ ws) {
  extern __shared__ float smem[];            // 131,200 B dynamic LDS
  float* w_lds = smem;                       // NCOUT * FWSTR
  float* x_buf = smem + NCOUT * FWSTR;       // 2 * XSLAB

  int tid  = threadIdx.x;
  int lane = tid & 31;
  int wave = tid >> 5;
  int half = lane >> 4;
  int nm   = lane & 15;

  int bx   = blockIdx.x;            // (b*50 + t)*8 + hblk
  int hblk = bx & 7;
  int bt   = bx >> 3;
  int t    = bt % TT;
  int b    = bt / TT;
  int h0   = hblk * 4;

  int h_local = wave >> 1;          // 0..3
  int wb16    = (wave & 1) * 16;
  int w_g     = wb16 + nm;          // 0..31

  v8f acc[4];
  #pragma unroll
  for (int ct = 0; ct < 4; ++ct)
    #pragma unroll
    for (int i = 0; i < 8; ++i) acc[ct][i] = 0.f;

  // one-time full-weight staging, permuted to k' (coalesced global reads)
  #pragma unroll 1
  for (int li = tid; li < NCOUT * KTOT; li += 256) {
    int cc  = li / KTOT;
    int kk  = li % KTOT;            // original order: cin*27 + q
    int kc2 = kk / KCH;
    int rem = kk % KCH;
    int cl  = rem / 27;
    int q   = rem % 27;
    w_lds[cc * FWSTR + kc2 * KCH + q * 4 + cl] = w[li];
  }
  stage_x_async(x, x_buf, b, t, h0, 0, tid);
  asm volatile("s_wait_asynccnt 0x0" ::: "memory");
  __syncthreads();

  #pragma unroll 1
  for (int kc = 0; kc < 4; ++kc) {
    float* cur = x_buf + (kc & 1) * XSLAB;
    if (kc < 3)
      stage_x_async(x, x_buf + ((kc + 1) & 1) * XSLAB, b, t, h0, kc + 1, tid);

    // hoisted bases: 3 x-planes (per kd), 4 weight rows (per ct)
    const float* xb0 = &cur[(h_local * 34 + w_g) * 4 + half * 2];
    const float* xb1 = xb0 + XPLANE;
    const float* xb2 = xb1 + XPLANE;
    const float* wp0 = &w_lds[nm * FWSTR + kc * KCH + half * 2];
    const float* wp1 = wp0 + 16 * FWSTR;
    const float* wp2 = wp1 + 16 * FWSTR;
    const float* wp3 = wp2 + 16 * FWSTR;

    #pragma unroll
    for (int q = 0; q < 27; ++q) {
      const int kd = q / 9, kh = (q % 9) / 3, kw = q % 3;   // constants after unroll
      const float* xp = (kd == 0) ? xb0 : ((kd == 1) ? xb1 : xb2);
      v2f bf = *(const v2f*)&xp[(kh * 34 + kw) * 4];
      v2f a0 = *(const v2f*)&wp0[q * 4];
      v2f a1 = *(const v2f*)&wp1[q * 4];
      v2f a2 = *(const v2f*)&wp2[q * 4];
      v2f a3 = *(const v2f*)&wp3[q * 4];
      acc[0] = __builtin_amdgcn_wmma_f32_16x16x4_f32(false, a0, false, bf, (short)0, acc[0], false, false);
      acc[1] = __builtin_amdgcn_wmma_f32_16x16x4_f32(false, a1, false, bf, (short)0, acc[1], false, false);
      acc[2] = __builtin_amdgcn_wmma_f32_16x16x4_f32(false, a2, false, bf, (short)0, acc[2], false, false);
      acc[3] = __builtin_amdgcn_wmma_f32_16x16x4_f32(false, a3, false, bf, (short)0, acc[3], false, false);
    }
    asm volatile("s_wait_asynccnt 0x0" ::: "memory");
    __syncthreads();
  }

  float scale = 1.0f - beta[0];
  int pos = (h0 + h_local) * NW + w_g;
  float* conv_ws = ws + WS_CONV;
  #pragma unroll
  for (int ct = 0; ct < 4; ++ct)
    #pragma unroll
    for (int i = 0; i < 8; ++i) {
      int cout = ct * 16 + half * 8 + i;
      conv_ws[(((size_t)b * TT + t) * NCOUT + cout) * HWPOS + pos] = acc[ct][i] * scale;
    }
}

// ---------------- per-timestep: spk@M^T GEMM + membrane update + threshold ---
__global__ __launch_bounds__(256) void step_kernel(const float* __restrict__ beta,
                                                   const float* __restrict__ bias,
                                                   float* __restrict__ ws,
                                                   float* __restrict__ out, int t) {
  __shared__ float spk_lds[128 * SSTR];   // 34,816 B
  __shared__ float m_lds[NCOUT * SSTR];   // 17,408 B (also reused for reduction)

  int tid  = threadIdx.x;
  int lane = tid & 31;
  int wave = tid >> 5;
  int half = lane >> 4;
  int nm   = lane & 15;

  int bx   = blockIdx.x;     // b*8 + pblk
  int b    = bx >> 3;
  int pos0 = (bx & 7) * 128;

  const float* mmt  = ws + WS_MMT;
  const float* invn = ws + WS_INV;
  float* mem     = ws + WS_MEM;
  float* tsum    = ws + WS_TSUM;
  float* conv_ws = ws + WS_CONV;
  float bet = beta[0];

  v8f acc[4];
  #pragma unroll
  for (int ct = 0; ct < 4; ++ct)
    #pragma unroll
    for (int i = 0; i < 8; ++i) acc[ct][i] = 0.f;

  if (t > 0) {
    #pragma unroll 1
    for (int li = tid; li < NCOUT * NCOUT; li += 256) {
      int c = li >> 6, k = li & 63;
      m_lds[c * SSTR + k] = mmt[li];
    }
    #pragma unroll 1
    for (int li = tid; li < 128 * NCOUT; li += 256) {
      int c = li >> 7, p = li & 127;
      spk_lds[p * SSTR + c] =
          out[(((size_t)b * NCOUT + c) * TT + (t - 1)) * HWPOS + pos0 + p];
    }
    __syncthreads();

    const float* sbase = &spk_lds[(wave * 16 + nm) * SSTR + half * 2];
    const float* mp0 = &m_lds[nm * SSTR + half * 2];
    const float* mp1 = mp0 + 16 * SSTR;
    const float* mp2 = mp1 + 16 * SSTR;
    const float* mp3 = mp2 + 16 * SSTR;
    #pragma unroll
    for (int k0 = 0; k0 < NCOUT; k0 += 4) {
      v2f bf = *(const v2f*)&sbase[k0];
      v2f a0 = *(const v2f*)&mp0[k0];
      v2f a1 = *(const v2f*)&mp1[k0];
      v2f a2 = *(const v2f*)&mp2[k0];
      v2f a3 = *(const v2f*)&mp3[k0];
      acc[0] = __builtin_amdgcn_wmma_f32_16x16x4_f32(false, a0, false, bf, (short)0, acc[0], false, false);
      acc[1] = __builtin_amdgcn_wmma_f32_16x16x4_f32(false, a1, false, bf, (short)0, acc[1], false, false);
      acc[2] = __builtin_amdgcn_wmma_f32_16x16x4_f32(false, a2, false, bf, (short)0, acc[2], false, false);
      acc[3] = __builtin_amdgcn_wmma_f32_16x16x4_f32(false, a3, false, bf, (short)0, acc[3], false, false);
    }
  }

  int posg = pos0 + wave * 16 + nm;
  float local = 0.f;
  #pragma unroll
  for (int ct = 0; ct < 4; ++ct)
    #pragma unroll
    for (int i = 0; i < 8; ++i) {
      int cout = ct * 16 + half * 8 + i;
      size_t mi = ((size_t)b * NCOUT + cout) * HWPOS + posg;
      float mold = (t > 0) ? mem[mi] : 0.f;
      float mnew = bet * mold + acc[ct][i] +
                   conv_ws[(((size_t)b * TT + t) * NCOUT + cout) * HWPOS + posg];
      mem[mi] = mnew;
      float mthr = mnew * invn[cout] - bias[cout];
      float s = (mthr > 0.f) ? 1.f : 0.f;
      out[(((size_t)b * NCOUT + cout) * TT + t) * HWPOS + posg] = s;
      local += s;
    }

  __syncthreads();
  m_lds[tid] = local;
  __syncthreads();
  for (int off = 128; off > 0; off >>= 1) {
    if (tid < off) m_lds[tid] += m_lds[tid + off];
    __syncthreads();
  }
  if (tid == 0) atomicAdd(&tsum[t], m_lds[0]);
}

// ---------------- finalize scalars -------------------------------------------
__global__ void finalize_kernel(float* __restrict__ ws, float* __restrict__ out) {
  if (threadIdx.x == 0) {
    const float* ts = ws + WS_TSUM;
    float tot = 0.f, mx = -1.f;
    for (int t = 0; t < TT; ++t) {
      tot += ts[t];
      mx = fmaxf(mx, ts[t]);
    }
    out[NOUT_SPK]     = 0.5f * tot / 52428800.0f;  // 0.5*mean(spk^2), spk binary
    out[NOUT_SPK + 1] = mx / 1048576.0f;           // max_t sum / (B*C*H*W)
  }
}

extern "C" void kernel_launch(void* const* d_in, const int* in_sizes, int n_in,
                              void* d_out, int out_size, void* d_ws, size_t ws_size,
                              hipStream_t stream) {
  const float* x    = (const float*)d_in[0];
  const float* w    = (const float*)d_in[1];
  const float* v    = (const float*)d_in[2];
  const float* beta = (const float*)d_in[3];
  const float* bias = (const float*)d_in[4];
  // d_in[5] = sigma: only used in backward; forward is a hard threshold.
  float* out = (float*)d_out;
  float* ws  = (float*)d_ws;

  size_t conv_lds = (size_t)(NCOUT * FWSTR + 2 * XSLAB) * sizeof(float); // 131,200 B

  hipLaunchKernelGGL(prep_kernel, dim3(1), dim3(256), 0, stream, w, v, beta, ws);
  hipLaunchKernelGGL(conv_kernel, dim3(6400), dim3(256), conv_lds, stream, x, w, beta, ws);
  for (int t = 0; t < TT; ++t)
    hipLaunchKernelGGL(step_kernel, dim3(128), dim3(256), 0, stream, beta, bias, ws, out, t);
  hipLaunchKernelGGL(finalize_kernel, dim3(1), dim3(1), 0, stream, ws, out);
}